// FocalLoss_20976620273983
// MI455X (gfx1250) — compile-verified
//
#include <hip/hip_runtime.h>

// Problem constants (match reference).
#define NB 8          // batch
#define NA 150000     // anchors
#define NC 10         // classes
#define TPB 256       // threads per block (8 wave32)
#define APT 4         // anchors per thread
#define TILE (TPB * APT)                  // 1024 anchors per block
#define NBLK ((NA + TILE - 1) / TILE)     // 147 tiles per image

typedef float v2f __attribute__((ext_vector_type(2)));
typedef float v8f __attribute__((ext_vector_type(8)));

// Sum a float across all 32 lanes of the wave using the CDNA5 matrix pipe.
// A-matrix (16x4 f32, 2 VGPRs/lane): VGPR0 = partial, VGPR1 = 0.
// B = ones(4x16). D[m][n] = rowsum(m) = partial[m] + partial[m+16].
// Summing D's 8 accumulator regs gives half-wave sums; one xor-16 shuffle
// completes the full 32-lane sum. EXEC is all-1s here (called uniformly).
__device__ __forceinline__ float wave_sum32(float x) {
    v2f a;  a[0] = x;    a[1] = 0.0f;
    v2f bo; bo[0] = 1.0f; bo[1] = 1.0f;
    v8f c = {};
    v8f d = __builtin_amdgcn_wmma_f32_16x16x4_f32(
        /*neg_a=*/false, a, /*neg_b=*/false, bo,
        /*c_mod=*/(short)0, c, /*reuse_a=*/false, /*reuse_b=*/false);
    float s = d[0] + d[1] + d[2] + d[3] + d[4] + d[5] + d[6] + d[7];
    s += __shfl_xor(s, 16, 32);
    return s;
}

// Stage 1: per-(image, anchor-tile) partial sums of (att_loss, reg_loss, num_pos).
__global__ void __launch_bounds__(TPB)
focal_partial_kernel(const float* __restrict__ attributes,   // (B, A, C)
                     const float* __restrict__ regressions,  // (B, A, 4)
                     const float* __restrict__ anchors,      // (1, A, 4)
                     const float* __restrict__ annotations,  // (B, 14)
                     float* __restrict__ ws)                 // (B, NBLK, 3)
{
    const int b   = blockIdx.y;
    const int tid = threadIdx.x;

    // Per-image ground truth (uniform address -> scalar loads / L2 broadcast).
    const float* ann = annotations + b * (4 + NC);
    const float bx0 = ann[0], by0 = ann[1], bx1 = ann[2], by1 = ann[3];
    float lab[NC];
#pragma unroll
    for (int c = 0; c < NC; ++c) lab[c] = ann[4 + c];

    const float gw_raw = bx1 - bx0;
    const float gh_raw = by1 - by0;
    const float area   = gw_raw * gh_raw;
    const float gcx    = bx0 + 0.5f * gw_raw;
    const float gcy    = by0 + 0.5f * gh_raw;
    const float lgw    = __logf(fmaxf(gw_raw, 1.0f));
    const float lgh    = __logf(fmaxf(gh_raw, 1.0f));

    float attS = 0.0f, rlS = 0.0f, npS = 0.0f;

#pragma unroll
    for (int k = 0; k < APT; ++k) {
        const int i = blockIdx.x * TILE + k * TPB + tid;
        if (i < NA) {
            const float4 an = *reinterpret_cast<const float4*>(anchors + 4 * (size_t)i);
            const float aw  = an.z - an.x;
            const float ah  = an.w - an.y;
            const float acx = an.x + 0.5f * aw;
            const float acy = an.y + 0.5f * ah;

            const float iw    = fmaxf(fminf(an.z, bx1) - fmaxf(an.x, bx0), 0.0f);
            const float ih    = fmaxf(fminf(an.w, by1) - fmaxf(an.y, by0), 0.0f);
            const float inter = iw * ih;
            const float uni   = fmaxf(aw * ah + area - inter, 1e-8f);
            const float iou   = inter / uni;

            const bool neg = iou < 0.4f;
            const bool pos = iou >= 0.5f;

            if (pos || neg) {
                // Focal loss over 10 classes. targets are exactly 0/1 here;
                // ignored anchors (0.4 <= iou < 0.5) were skipped entirely.
                const float* ap = attributes + ((size_t)b * NA + i) * NC;
#pragma unroll
                for (int c = 0; c < NC; ++c) {
                    const float p   = fminf(fmaxf(ap[c], 1e-4f), 0.9999f);
                    const bool one  = pos && (lab[c] == 1.0f);
                    const float q   = one ? (1.0f - p) : p;          // focal weight base
                    const float lp  = __logf(one ? p : (1.0f - p));  // log of predicted prob of target
                    const float af  = one ? 0.25f : 0.75f;
                    attS += af * q * q * (-lp);
                }
            }

            if (pos) {
                npS += 1.0f;
                const float4 r = *reinterpret_cast<const float4*>(
                    regressions + ((size_t)b * NA + i) * 4);
                // Targets / STD  (STD = [0.1, 0.1, 0.2, 0.2])
                const float tx = (gcx - acx) / aw * 10.0f;
                const float ty = (gcy - acy) / ah * 10.0f;
                const float tw = (lgw - __logf(aw)) * 5.0f;
                const float th = (lgh - __logf(ah)) * 5.0f;
                const float dif[4] = { fabsf(tx - r.x), fabsf(ty - r.y),
                                       fabsf(tw - r.z), fabsf(th - r.w) };
#pragma unroll
                for (int j = 0; j < 4; ++j) {
                    const float dd = dif[j];
                    rlS += (dd <= (1.0f / 9.0f)) ? 4.5f * dd * dd
                                                 : dd - (0.5f / 9.0f);
                }
            }
        }
    }

    // Wave32 reduction via the matrix pipe, then 8 waves -> block via LDS.
    const float wa = wave_sum32(attS);
    const float wr = wave_sum32(rlS);
    const float wn = wave_sum32(npS);

    __shared__ float sm[(TPB / 32) * 3];
    const int wave = tid >> 5;
    const int lane = tid & 31;
    if (lane == 0) {
        sm[wave * 3 + 0] = wa;
        sm[wave * 3 + 1] = wr;
        sm[wave * 3 + 2] = wn;
    }
    __syncthreads();
    if (tid == 0) {
        float a = 0.0f, r = 0.0f, n = 0.0f;
#pragma unroll
        for (int w = 0; w < TPB / 32; ++w) {
            a += sm[w * 3 + 0];
            r += sm[w * 3 + 1];
            n += sm[w * 3 + 2];
        }
        const size_t idx = ((size_t)b * gridDim.x + blockIdx.x) * 3;
        ws[idx + 0] = a;
        ws[idx + 1] = r;
        ws[idx + 2] = n;
    }
}

// Stage 2: reduce per-block partials per image, normalize, write the 2 means.
__global__ void __launch_bounds__(TPB)
focal_final_kernel(const float* __restrict__ ws, float* __restrict__ out)
{
    const int tid = threadIdx.x;
    __shared__ float sm[(TPB / 32) * 3];
    __shared__ float acc2[2];

    float att_acc = 0.0f, reg_acc = 0.0f;

    for (int b = 0; b < NB; ++b) {
        float a = 0.0f, r = 0.0f, n = 0.0f;
        for (int j = tid; j < NBLK; j += TPB) {
            const size_t idx = ((size_t)b * NBLK + j) * 3;
            a += ws[idx + 0];
            r += ws[idx + 1];
            n += ws[idx + 2];
        }
        const float wa = wave_sum32(a);
        const float wr = wave_sum32(r);
        const float wn = wave_sum32(n);
        const int wave = tid >> 5, lane = tid & 31;
        if (lane == 0) {
            sm[wave * 3 + 0] = wa;
            sm[wave * 3 + 1] = wr;
            sm[wave * 3 + 2] = wn;
        }
        __syncthreads();
        if (tid == 0) {
            float A_ = 0.0f, R_ = 0.0f, N_ = 0.0f;
#pragma unroll
            for (int w = 0; w < TPB / 32; ++w) {
                A_ += sm[w * 3 + 0];
                R_ += sm[w * 3 + 1];
                N_ += sm[w * 3 + 2];
            }
            att_acc += A_ / fmaxf(N_, 1.0f);
            reg_acc += (N_ > 0.0f) ? R_ / (4.0f * N_) : 0.0f;
        }
        __syncthreads();  // sm reused next image
    }

    if (tid == 0) {
        acc2[0] = att_acc * (1.0f / NB);
        acc2[1] = reg_acc * (1.0f / NB);
        out[0] = acc2[0];
        out[1] = acc2[1];
    }
}

extern "C" void kernel_launch(void* const* d_in, const int* in_sizes, int n_in,
                              void* d_out, int out_size, void* d_ws, size_t ws_size,
                              hipStream_t stream) {
    const float* attributes  = (const float*)d_in[0];  // (B, A, C)
    const float* regressions = (const float*)d_in[1];  // (B, A, 4)
    const float* anchors     = (const float*)d_in[2];  // (1, A, 4)
    const float* annotations = (const float*)d_in[3];  // (B, 14)
    float* out = (float*)d_out;                        // [att_mean, reg_mean]
    float* ws  = (float*)d_ws;                         // (B, NBLK, 3) floats

    dim3 grid(NBLK, NB, 1);
    focal_partial_kernel<<<grid, TPB, 0, stream>>>(attributes, regressions,
                                                   anchors, annotations, ws);
    focal_final_kernel<<<1, TPB, 0, stream>>>(ws, out);
}